// MG_GATs_82703890252227
// MI455X (gfx1250) — compile-verified
//
#include <hip/hip_runtime.h>
#include <hip/hip_bf16.h>
#include <math.h>

// ---------------- problem constants ----------------
#define NNODES   20480      // 64*5*64
#define TDIM     128
#define FDIM     256        // HEADS*HID
#define HEADS    4
#define HID      64
#define NEDGE    327680
#define NTOT     (NEDGE + NNODES)   // with self loops
#define NGRAPH   64
#define NCLS     4
#define NEGSLOPE 0.2f

typedef float v2f __attribute__((ext_vector_type(2)));
typedef float v8f __attribute__((ext_vector_type(8)));

// ---------------- small helpers ----------------
__device__ __forceinline__ void atomicMaxF(float* addr, float v) {
    if (v >= 0.0f) atomicMax((int*)addr, __float_as_int(v));
    else           atomicMin((unsigned int*)addr, __float_as_uint(v));
}
__device__ __forceinline__ void atomAddF(float* addr, float v) {
    unsafeAtomicAdd(addr, v);   // global_atomic_add_f32
}
__device__ __forceinline__ float leakyf(float x) {
    return x >= 0.0f ? x : NEGSLOPE * x;
}

// ---------------- generic fill ----------------
__global__ void fill_kernel(float* __restrict__ p, float v, int n) {
    int i = blockIdx.x * blockDim.x + threadIdx.x;
    for (; i < n; i += gridDim.x * blockDim.x) p[i] = v;
}

// ---------------- W -> Wt transpose (K x N  ->  N x K) ----------------
__global__ void transpose_kernel(const float* __restrict__ W, float* __restrict__ Wt,
                                 int K, int Ncols) {
    int i = blockIdx.x * blockDim.x + threadIdx.x;
    int tot = K * Ncols;
    for (; i < tot; i += gridDim.x * blockDim.x) {
        int k = i / Ncols, n = i - k * Ncols;
        Wt[(size_t)n * K + k] = W[i];
    }
}

// ---------------- fp32 WMMA GEMM : C[20480,256] = A[20480,K] * Wt^T ----------------
// Wt stored [256][K] (transposed weights, L2-resident). Each wave computes a
// 16-row x 128-col tile (8 N-tiles, 64 accum VGPRs); 2 waves cover the full
// 256 columns of a row stripe. Per k-step all 9 fragments (1 A + 8 B) are
// batch-loaded before the 8 WMMAs so the loads coalesce under one wait and
// the k-loop can be software-pipelined.
__global__ __launch_bounds__(256) void gemm_wmma_f32(const float* __restrict__ A,
                                                     const float* __restrict__ Wt,
                                                     float* __restrict__ C, int K) {
    int wid     = threadIdx.x >> 5;               // 0..7
    int wave    = blockIdx.x * 8 + wid;
    int rowtile = wave >> 1;                      // 16-row stripe
    int nhalf   = wave & 1;                       // which 128 columns
    int rowbase = rowtile * 16;
    if (rowbase >= NNODES) return;
    int lane = threadIdx.x & 31;
    int half = lane >> 4;     // 0: K+0/K+1 group, 1: K+2/K+3 group
    int lid  = lane & 15;

    v8f acc[8];
#pragma unroll
    for (int t = 0; t < 8; ++t) acc[t] = (v8f){0.f,0.f,0.f,0.f,0.f,0.f,0.f,0.f};

    const float* aptr  = A  + (size_t)(rowbase + lid) * K + 2 * half;
    const float* bbase = Wt + (size_t)(nhalf * 128 + lid) * K + 2 * half;
    const size_t tstride = (size_t)16 * K;        // floats between N-tiles

#pragma unroll 2
    for (int k = 0; k < K; k += 4) {
        v2f a = *(const v2f*)(aptr + k);          // A frag: lane=M, {k+2h, k+2h+1}
        v2f b[8];
#pragma unroll
        for (int t = 0; t < 8; ++t)               // batch all B-frag loads
            b[t] = *(const v2f*)(bbase + t * tstride + k);
#pragma unroll
        for (int t = 0; t < 8; ++t)
            acc[t] = __builtin_amdgcn_wmma_f32_16x16x4_f32(
                false, a, false, b[t], (short)0, acc[t], false, false);
    }

    // C/D layout: VGPR j -> rows rowbase+j (lanes 0-15) / rowbase+j+8 (lanes 16-31)
#pragma unroll
    for (int t = 0; t < 8; ++t) {
        int col = nhalf * 128 + t * 16 + lid;
#pragma unroll
        for (int j = 0; j < 8; ++j) {
            int row = rowbase + j + 8 * half;
            C[(size_t)row * FDIM + col] = acc[t][j];
        }
    }
}

// ---------------- attention coefficients: asrc/adst [N,4] ----------------
__global__ void attn_kernel(const float* __restrict__ H,
                            const float* __restrict__ avs,   // [256] flattened a_src
                            const float* __restrict__ avd,   // [256] flattened a_dst
                            float* __restrict__ asrc, float* __restrict__ adst) {
    __shared__ float ss[256];
    __shared__ float sd[256];
    int n = blockIdx.x;
    int t = threadIdx.x;
    float h = H[(size_t)n * FDIM + t];
    ss[t] = h * avs[t];
    sd[t] = h * avd[t];
    __syncthreads();
    for (int off = 32; off > 0; off >>= 1) {
        if ((t & 63) < off) { ss[t] += ss[t + off]; sd[t] += sd[t + off]; }
        __syncthreads();
    }
    if ((t & 63) == 0) {
        asrc[n * HEADS + (t >> 6)] = ss[t];
        adst[n * HEADS + (t >> 6)] = sd[t];
    }
}

// ---------------- edge pass 1: segment max ----------------
__global__ void edge_max_kernel(const int* __restrict__ src, const int* __restrict__ dst,
                                const float* __restrict__ asrc, const float* __restrict__ adst,
                                float* __restrict__ mmax) {
    int i = blockIdx.x * blockDim.x + threadIdx.x;
    if (i >= NTOT) return;
    int s0 = (i < NEDGE) ? src[i] : (i - NEDGE);
    int d0 = (i < NEDGE) ? dst[i] : (i - NEDGE);
#pragma unroll
    for (int h = 0; h < HEADS; ++h) {
        float e = leakyf(asrc[s0 * HEADS + h] + adst[d0 * HEADS + h]);
        atomicMaxF(mmax + d0 * HEADS + h, e);
    }
}

// ---------------- edge pass 2: exp + segment sum ----------------
__global__ void edge_expsum_kernel(const int* __restrict__ src, const int* __restrict__ dst,
                                   const float* __restrict__ asrc, const float* __restrict__ adst,
                                   const float* __restrict__ mmax, float* __restrict__ ssum) {
    int i = blockIdx.x * blockDim.x + threadIdx.x;
    if (i >= NTOT) return;
    int s0 = (i < NEDGE) ? src[i] : (i - NEDGE);
    int d0 = (i < NEDGE) ? dst[i] : (i - NEDGE);
#pragma unroll
    for (int h = 0; h < HEADS; ++h) {
        float e = leakyf(asrc[s0 * HEADS + h] + adst[d0 * HEADS + h]);
        float e2 = __expf(e - mmax[d0 * HEADS + h]);
        atomAddF(ssum + d0 * HEADS + h, e2);
    }
}

// ---------------- edge pass 3: alpha * h[src] scattered to dst ----------------
// 64 lanes per edge, each lane owns 4 contiguous channels (one head per 16 lanes)
__global__ void edge_scatter_kernel(const int* __restrict__ src, const int* __restrict__ dst,
                                    const float* __restrict__ asrc, const float* __restrict__ adst,
                                    const float* __restrict__ mmax, const float* __restrict__ ssum,
                                    const float* __restrict__ Hin, float* __restrict__ Hout) {
    int t   = threadIdx.x;              // 256
    int grp = blockIdx.x * 4 + (t >> 6);
    if (grp >= NTOT) return;
    int l64 = t & 63;
    int s0 = (grp < NEDGE) ? src[grp] : (grp - NEDGE);
    int d0 = (grp < NEDGE) ? dst[grp] : (grp - NEDGE);
    int h  = l64 >> 4;
    float e  = leakyf(asrc[s0 * HEADS + h] + adst[d0 * HEADS + h]);
    float al = __expf(e - mmax[d0 * HEADS + h]) / (ssum[d0 * HEADS + h] + 1e-16f);
    int c0 = l64 * 4;
    float4 hv = *(const float4*)(Hin + (size_t)s0 * FDIM + c0);
    float* op = Hout + (size_t)d0 * FDIM + c0;
    atomAddF(op + 0, al * hv.x);
    atomAddF(op + 1, al * hv.y);
    atomAddF(op + 2, al * hv.z);
    atomAddF(op + 3, al * hv.w);
}

// ---------------- per-feature sum / sumsq of (H + bias) ----------------
#define STAT_ROWS 128
__global__ void stats_kernel(const float* __restrict__ H, const float* __restrict__ bias,
                             float* __restrict__ stats) {
    int f  = threadIdx.x;          // 256 features
    float b = bias[f];
    int r0 = blockIdx.x * STAT_ROWS;
    float s = 0.f, s2 = 0.f;
    for (int n = r0; n < r0 + STAT_ROWS; ++n) {
        float v = H[(size_t)n * FDIM + f] + b;
        s += v; s2 += v * v;
    }
    atomAddF(stats + f, s);
    atomAddF(stats + FDIM + f, s2);
}

// ---------------- GraphNorm + PReLU -> Out ----------------
__global__ void norm_act_kernel(const float* __restrict__ H, const float* __restrict__ bias,
                                const float* __restrict__ stats,
                                const float* __restrict__ gw, const float* __restrict__ gb,
                                const float* __restrict__ ga, const float* __restrict__ prelu,
                                float* __restrict__ Out) {
    int f = threadIdx.x;
    const float invn = 1.0f / (float)NNODES;
    float mean = stats[f] * invn;
    float ex2  = stats[FDIM + f] * invn;
    float a = ga[f];
    float var = ex2 - 2.f * a * mean * mean + a * a * mean * mean;
    float inv = rsqrtf(var + 1e-5f);
    float w = gw[f], bb = gb[f], b = bias[f], p = prelu[0];
    int r0 = blockIdx.x * STAT_ROWS;
    for (int n = r0; n < r0 + STAT_ROWS; ++n) {
        float v = H[(size_t)n * FDIM + f] + b;
        float y = w * (v - a * mean) * inv + bb;
        Out[(size_t)n * FDIM + f] = (y >= 0.f) ? y : p * y;
    }
}

// ---------------- GraphNorm + PReLU + mean-pool accumulate ----------------
__global__ void norm_pool_kernel(const float* __restrict__ H, const float* __restrict__ bias,
                                 const float* __restrict__ stats,
                                 const float* __restrict__ gw, const float* __restrict__ gb,
                                 const float* __restrict__ ga, const float* __restrict__ prelu,
                                 const int* __restrict__ batch,
                                 float* __restrict__ gpool, float* __restrict__ gcnt) {
    int f = threadIdx.x;
    const float invn = 1.0f / (float)NNODES;
    float mean = stats[f] * invn;
    float ex2  = stats[FDIM + f] * invn;
    float a = ga[f];
    float var = ex2 - 2.f * a * mean * mean + a * a * mean * mean;
    float inv = rsqrtf(var + 1e-5f);
    float w = gw[f], bb = gb[f], b = bias[f], p = prelu[0];
    int r0 = blockIdx.x * STAT_ROWS;
    for (int n = r0; n < r0 + STAT_ROWS; ++n) {
        int bg = batch[n];
        float v = H[(size_t)n * FDIM + f] + b;
        float y = w * (v - a * mean) * inv + bb;
        y = (y >= 0.f) ? y : p * y;
        atomAddF(gpool + (size_t)bg * FDIM + f, y);
        if (f == 0) atomAddF(gcnt + bg, 1.0f);
    }
}

// ---------------- Conv1d(1,1,4,stride4) + AvgPool1d(4) ----------------
__global__ void down_kernel(const float* __restrict__ g, const float* __restrict__ cnt,
                            const float* __restrict__ cw, const float* __restrict__ cb,
                            float* __restrict__ out, int class_off) {
    int t = blockIdx.x * blockDim.x + threadIdx.x;
    if (t >= NGRAPH * 16) return;
    int gi = t >> 4, k = t & 15;
    float c = cnt[gi]; if (c < 1.f) c = 1.f;
    float inv = 1.f / c;
    float w0 = cw[0], w1 = cw[1], w2 = cw[2], w3 = cw[3], b = cb[0];
    float acc = 0.f;
    for (int j = 4 * k; j < 4 * k + 4; ++j) {
        const float* gp = g + (size_t)gi * FDIM + j * 4;
        acc += (gp[0] * w0 + gp[1] * w1 + gp[2] * w2 + gp[3] * w3) * inv + b;
    }
    out[gi * (NCLS * 16) + class_off + k] = acc * 0.25f;
}

// ---------------- host orchestration ----------------
extern "C" void kernel_launch(void* const* d_in, const int* in_sizes, int n_in,
                              void* d_out, int out_size, void* d_ws, size_t ws_size,
                              hipStream_t stream) {
    const float* x      = (const float*)d_in[0];
    const int*   ei     = (const int*)  d_in[1];
    const int*   batch  = (const int*)  d_in[2];
    const float* W1     = (const float*)d_in[3];
    const float* asr1   = (const float*)d_in[4];
    const float* ads1   = (const float*)d_in[5];
    const float* b1     = (const float*)d_in[6];
    const float* W2     = (const float*)d_in[7];
    const float* asr2   = (const float*)d_in[8];
    const float* ads2   = (const float*)d_in[9];
    const float* b2     = (const float*)d_in[10];
    const float* gw1    = (const float*)d_in[11];
    const float* gb1    = (const float*)d_in[12];
    const float* ga1    = (const float*)d_in[13];
    const float* gw2    = (const float*)d_in[14];
    const float* gb2    = (const float*)d_in[15];
    const float* ga2    = (const float*)d_in[16];
    const float* pr1    = (const float*)d_in[17];
    const float* pr2    = (const float*)d_in[18];
    const float* cw     = (const float*)d_in[19];
    const float* cb     = (const float*)d_in[20];
    float* out = (float*)d_out;

    // workspace layout (floats)
    float* ws    = (float*)d_ws;
    float* bufA  = ws;                         // 20480*256
    float* bufB  = bufA + (size_t)NNODES * FDIM;
    float* Wt    = bufB + (size_t)NNODES * FDIM;   // 256*256 max
    float* asrc  = Wt + 256 * 256;
    float* adst  = asrc + NNODES * HEADS;
    float* mmax  = adst + NNODES * HEADS;
    float* ssum  = mmax + NNODES * HEADS;
    float* stats = ssum + NNODES * HEADS;      // 512
    float* gpool = stats + 2 * FDIM;           // 64*256
    float* gcnt  = gpool + NGRAPH * FDIM;      // 64

    const float NEGINF = -__builtin_huge_valf();
    const int EDGE_GRID  = (NTOT + 255) / 256;
    const int SCAT_GRID  = (NTOT + 3) / 4;
    const int GEMM_GRID  = (NNODES / 16) * 2 / 8;  // 2 waves per row stripe, 8 waves/block
    const int NF_BIG     = NNODES * FDIM;

    for (int c = 0; c < NCLS; ++c) {
        const int*   srcp = ei + (size_t)c * 2 * NEDGE;
        const int*   dstp = srcp + NEDGE;
        const int*   bat  = batch + (size_t)c * NNODES;

        // ---------- layer 1 ----------
        transpose_kernel<<<128, 256, 0, stream>>>(W1 + (size_t)c * TDIM * FDIM, Wt, TDIM, FDIM);
        fill_kernel<<<320, 256, 0, stream>>>(mmax, NEGINF, NNODES * HEADS);
        fill_kernel<<<320, 256, 0, stream>>>(ssum, 0.f, NNODES * HEADS);
        fill_kernel<<<2, 256, 0, stream>>>(stats, 0.f, 2 * FDIM);

        gemm_wmma_f32<<<GEMM_GRID, 256, 0, stream>>>(x, Wt, bufA, TDIM);
        attn_kernel<<<NNODES, 256, 0, stream>>>(bufA, asr1 + c * FDIM, ads1 + c * FDIM, asrc, adst);
        edge_max_kernel<<<EDGE_GRID, 256, 0, stream>>>(srcp, dstp, asrc, adst, mmax);
        edge_expsum_kernel<<<EDGE_GRID, 256, 0, stream>>>(srcp, dstp, asrc, adst, mmax, ssum);
        fill_kernel<<<2048, 256, 0, stream>>>(bufB, 0.f, NF_BIG);
        edge_scatter_kernel<<<SCAT_GRID, 256, 0, stream>>>(srcp, dstp, asrc, adst, mmax, ssum,
                                                           bufA, bufB);
        stats_kernel<<<NNODES / STAT_ROWS, 256, 0, stream>>>(bufB, b1 + c * FDIM, stats);
        norm_act_kernel<<<NNODES / STAT_ROWS, 256, 0, stream>>>(bufB, b1 + c * FDIM, stats,
            gw1 + c * FDIM, gb1 + c * FDIM, ga1 + c * FDIM, pr1 + c, bufA);

        // ---------- layer 2 ----------
        transpose_kernel<<<256, 256, 0, stream>>>(W2 + (size_t)c * FDIM * FDIM, Wt, FDIM, FDIM);
        fill_kernel<<<320, 256, 0, stream>>>(mmax, NEGINF, NNODES * HEADS);
        fill_kernel<<<320, 256, 0, stream>>>(ssum, 0.f, NNODES * HEADS);
        fill_kernel<<<2, 256, 0, stream>>>(stats, 0.f, 2 * FDIM);

        gemm_wmma_f32<<<GEMM_GRID, 256, 0, stream>>>(bufA, Wt, bufB, FDIM);
        attn_kernel<<<NNODES, 256, 0, stream>>>(bufB, asr2 + c * FDIM, ads2 + c * FDIM, asrc, adst);
        edge_max_kernel<<<EDGE_GRID, 256, 0, stream>>>(srcp, dstp, asrc, adst, mmax);
        edge_expsum_kernel<<<EDGE_GRID, 256, 0, stream>>>(srcp, dstp, asrc, adst, mmax, ssum);
        fill_kernel<<<2048, 256, 0, stream>>>(bufA, 0.f, NF_BIG);
        fill_kernel<<<64, 256, 0, stream>>>(gpool, 0.f, NGRAPH * FDIM);
        fill_kernel<<<1, 64, 0, stream>>>(gcnt, 0.f, NGRAPH);
        edge_scatter_kernel<<<SCAT_GRID, 256, 0, stream>>>(srcp, dstp, asrc, adst, mmax, ssum,
                                                           bufB, bufA);
        stats_kernel<<<NNODES / STAT_ROWS, 256, 0, stream>>>(bufA, b2 + c * FDIM, stats);
        norm_pool_kernel<<<NNODES / STAT_ROWS, 256, 0, stream>>>(bufA, b2 + c * FDIM, stats,
            gw2 + c * FDIM, gb2 + c * FDIM, ga2 + c * FDIM, pr2 + c, bat, gpool, gcnt);

        down_kernel<<<4, 256, 0, stream>>>(gpool, gcnt, cw + c * 4, cb + c, out, c * 16);
    }
}